// SageLayer_49331994362503
// MI455X (gfx1250) — compile-verified
//
#include <hip/hip_runtime.h>
#include <hip/hip_bf16.h>

typedef __attribute__((ext_vector_type(2))) float v2f;
typedef __attribute__((ext_vector_type(8))) float v8f;

#define N_NODES 50000
#define D_FEAT  128
#define D_OUT   128

// ---------------------------------------------------------------------------
// Kernel 0: zero the workspace (summed[N*128] + deg[N]) every call.
// ---------------------------------------------------------------------------
__global__ __launch_bounds__(256) void sage_zero_ws(float* ws, int n_elems) {
    int i = blockIdx.x * blockDim.x + threadIdx.x;
    if (i < n_elems) ws[i] = 0.0f;
}

// ---------------------------------------------------------------------------
// Kernel 1: edge scatter. One wave32 per edge; lane l handles feats [4l,4l+4).
// summed[dst] += features[src]; deg[dst] += 1.
// ---------------------------------------------------------------------------
__global__ __launch_bounds__(256) void sage_scatter(const int* __restrict__ src,
                                                    const int* __restrict__ dst,
                                                    const float* __restrict__ feat,
                                                    float* __restrict__ summed,
                                                    float* __restrict__ deg,
                                                    int n_edges) {
    int gtid = blockIdx.x * blockDim.x + threadIdx.x;
    int edge = gtid >> 5;           // one wave32 per edge
    int lane = gtid & 31;
    if (edge >= n_edges) return;

    int s = src[edge];
    int d = dst[edge];

    const float4* f4 = (const float4*)(feat + (size_t)s * D_FEAT);
    float4 v = f4[lane];            // 32 lanes x 4 floats = 128 feats

    float* out = summed + (size_t)d * D_FEAT + lane * 4;
    atomicAdd(out + 0, v.x);
    atomicAdd(out + 1, v.y);
    atomicAdd(out + 2, v.z);
    atomicAdd(out + 3, v.w);
    if (lane == 0) atomicAdd(deg + d, 1.0f);
}

// ---------------------------------------------------------------------------
// Kernel 2: fused mean-normalize + concat + GEMM + bias + ReLU via
// V_WMMA_F32_16X16X4_F32. One wave per 16x16 output tile; 8 waves per block
// cover the 8 column tiles (D_OUT=128) of one 16-node row tile.
//
// A fragment (16x4 f32, ISA 7.12.2): lane l holds row M=l%16,
//   VGPR0/1 = K = k0 + 2*(l>=16) + {0,1}.
// B fragment (4x16): lane l holds col N=l%16, same K striping (transpose of A).
// C/D (16x16 f32): VGPR v = row v + 8*(l>=16), col = l%16.
// ---------------------------------------------------------------------------
__global__ __launch_bounds__(256) void sage_wmma_gemm(const float* __restrict__ feat,
                                                      const float* __restrict__ summed,
                                                      const float* __restrict__ deg,
                                                      const float* __restrict__ weight,
                                                      const float* __restrict__ bias,
                                                      float* __restrict__ out) {
    const int wave = threadIdx.x >> 5;        // 0..7 -> column tile
    const int lane = threadIdx.x & 31;
    const int half = lane >> 4;               // 0: K+0/1, 1: K+2/3
    const int lq   = lane & 15;

    const int tile_row = blockIdx.x;          // 16 nodes per tile, 3125 tiles
    const int row = tile_row * 16 + lq;       // node this lane loads for A
    const int col = wave * 16 + lq;           // output column this lane loads for B

    const float inv = 1.0f / fmaxf(deg[row], 1.0f);

    const float* __restrict__ frow = feat   + (size_t)row * D_FEAT;
    const float* __restrict__ srow = summed + (size_t)row * D_FEAT;
    const float* __restrict__ wrow = weight + (size_t)col * (2 * D_FEAT);

    v8f c = {0.f, 0.f, 0.f, 0.f, 0.f, 0.f, 0.f, 0.f};

    #pragma unroll
    for (int k0 = 0; k0 < 2 * D_FEAT; k0 += 4) {
        const int k = k0 + 2 * half;          // this lane's K pair
        v2f a, b;
        if (k0 < D_FEAT) {                    // self features (k, k+1 < 128)
            a.x = frow[k];
            a.y = frow[k + 1];
        } else {                              // neighbor mean
            a.x = srow[k - D_FEAT] * inv;
            a.y = srow[k - D_FEAT + 1] * inv;
        }
        b.x = wrow[k];
        b.y = wrow[k + 1];
        // 8 args: (neg_a, A, neg_b, B, c_mod, C, reuse_a, reuse_b)
        c = __builtin_amdgcn_wmma_f32_16x16x4_f32(false, a, false, b,
                                                  (short)0, c, false, false);
    }

    const float bv = bias[col];
    #pragma unroll
    for (int v = 0; v < 8; ++v) {
        const int r = tile_row * 16 + v + 8 * half;
        out[(size_t)r * D_OUT + col] = fmaxf(c[v] + bv, 0.0f);
    }
}

// ---------------------------------------------------------------------------
// Launch. Inputs: nodes[i64->int, unused], edge_index[2*E], features[N*128],
// weight[128*256], bias[128]. Output: [N*128] f32.
// ---------------------------------------------------------------------------
extern "C" void kernel_launch(void* const* d_in, const int* in_sizes, int n_in,
                              void* d_out, int out_size, void* d_ws, size_t ws_size,
                              hipStream_t stream) {
    const int*   ei     = (const int*)d_in[1];
    const float* feat   = (const float*)d_in[2];
    const float* weight = (const float*)d_in[3];
    const float* bias   = (const float*)d_in[4];
    float*       out    = (float*)d_out;

    const int n_edges = in_sizes[1] / 2;
    const int* src = ei;
    const int* dst = ei + n_edges;

    float* summed = (float*)d_ws;                       // N*128 floats
    float* deg    = summed + (size_t)N_NODES * D_FEAT;  // N floats

    // 0) zero summed + deg (ws is not re-poisoned between replays)
    const int nz = N_NODES * D_FEAT + N_NODES;
    sage_zero_ws<<<(nz + 255) / 256, 256, 0, stream>>>((float*)d_ws, nz);

    // 1) edge scatter: one wave32 per edge, 8 edges per 256-thread block
    const int nthreads = n_edges * 32;
    sage_scatter<<<(nthreads + 255) / 256, 256, 0, stream>>>(src, dst, feat,
                                                             summed, deg, n_edges);

    // 2) fused normalize + concat-GEMM + bias + ReLU (WMMA fp32)
    sage_wmma_gemm<<<N_NODES / 16, 256, 0, stream>>>(feat, summed, deg,
                                                     weight, bias, out);
}